// LossFunction_29978871726169
// MI455X (gfx1250) — compile-verified
//
#include <hip/hip_runtime.h>
#include <stdint.h>

#define Nn 2048
#define Tt 2000
#define Kk 12
#define Dd 8
#define T1 1999

#define NT_TILE 32          // n per block tile
#define TT_TILE 40          // t per block tile  (2000/40 = 50, no ragged tiles)
#define TROW_DW 484         // LDS dwords per theta n-row: 40*12=480, pad to 484 (36 mod 64 -> conflict-free)

#define NCH 16              // time chunks for parallel scan
#define CHL 125             // chunk length (covers t=1..1999)

typedef __attribute__((ext_vector_type(2))) float v2f;
typedef __attribute__((ext_vector_type(8))) float v8f;

// ---------------------------------------------------------------------------
// Phase 1: eps[n,t] = y - theta0 - <theta[1:9], f[t]>  via V_WMMA_F32_16X16X4_F32
// theta tile streamed to LDS with global_load_async_to_lds_b128.
// Also repacks alpha/beta/omega into [t][n]-major planes for the scan phases.
// ---------------------------------------------------------------------------
__global__ __launch_bounds__(256) void garch_phase1(
    const float* __restrict__ y, const float* __restrict__ f,
    const float* __restrict__ theta,
    float* __restrict__ eps_out, float* __restrict__ aT,
    float* __restrict__ bT, float* __restrict__ oT,
    float* __restrict__ eps0)
{
    __shared__ float th_lds[NT_TILE * TROW_DW];   // 61,952 B
    __shared__ float f_lds[TT_TILE * Dd];         //  1,280 B

    const int tid  = threadIdx.x;
    const int lane = tid & 31;
    const int w    = tid >> 5;                    // wave id 0..7
    const int t0   = blockIdx.x * TT_TILE;
    const int n0   = blockIdx.y * NT_TILE;

    // stage f[t0..t0+39][0..7] (contiguous, coalesced)
    for (int i = tid; i < TT_TILE * Dd; i += 256)
        f_lds[i] = f[t0 * Dd + i];

    // async-stage theta: wave w owns n-rows 4w..4w+3; each row = 1920B = 120 x 16B
    for (int rr = 0; rr < 4; ++rr) {
        const int nrow = 4 * w + rr;
        const uint64_t gbase = (uint64_t)(uintptr_t)theta
                             + (uint64_t)(n0 + nrow) * (Tt * Kk * 4u)
                             + (uint64_t)t0 * (Kk * 4u);
        const uint32_t lbase = (uint32_t)(uintptr_t)&th_lds[nrow * TROW_DW];
        for (int i = lane; i < (TT_TILE * Kk) / 4; i += 32) {
            uint64_t ga = gbase + (uint64_t)i * 16u;
            uint32_t la = lbase + (uint32_t)i * 16u;
            asm volatile("global_load_async_to_lds_b128 %0, %1, off"
                         :: "v"(la), "v"(ga) : "memory");
        }
    }
    asm volatile("s_wait_asynccnt 0" ::: "memory");
    __syncthreads();

    const int h    = w & 1;                       // n half: rows 16h..16h+15
    const int koff = (lane < 16) ? 0 : 2;         // WMMA K split across lane halves
    const int nl   = 16 * h + (lane & 15);        // this lane's series within tile
    const float* row = &th_lds[nl * TROW_DW];     // lane's theta row (also B source)

    for (int r = 0; r < 5; ++r) {
        const int p  = (w >> 1) + 4 * r;          // t-pair 0..19
        const int ta = 2 * p;
        const int tb = ta + 1;

        // A: every row of A equals f[t]  (lanes<16 hold K=0,1; lanes>=16 hold K=2,3)
        v2f Aa0, Aa1, Ab0, Ab1;
        Aa0[0] = f_lds[ta * Dd + koff + 0]; Aa0[1] = f_lds[ta * Dd + koff + 1];
        Aa1[0] = f_lds[ta * Dd + koff + 4]; Aa1[1] = f_lds[ta * Dd + koff + 5];
        Ab0[0] = f_lds[tb * Dd + koff + 0]; Ab0[1] = f_lds[tb * Dd + koff + 1];
        Ab1[0] = f_lds[tb * Dd + koff + 4]; Ab1[1] = f_lds[tb * Dd + koff + 5];

        // B[k][n] = theta[n0+16h+n, t, 1+g+k]
        v2f Ba0, Ba1, Bb0, Bb1;
        Ba0[0] = row[ta * Kk + 1 + koff]; Ba0[1] = row[ta * Kk + 2 + koff];
        Ba1[0] = row[ta * Kk + 5 + koff]; Ba1[1] = row[ta * Kk + 6 + koff];
        Bb0[0] = row[tb * Kk + 1 + koff]; Bb0[1] = row[tb * Kk + 2 + koff];
        Bb1[0] = row[tb * Kk + 5 + koff]; Bb1[1] = row[tb * Kk + 6 + koff];

        v8f ca = {};
        ca = __builtin_amdgcn_wmma_f32_16x16x4_f32(false, Aa0, false, Ba0, (short)0, ca, false, false);
        ca = __builtin_amdgcn_wmma_f32_16x16x4_f32(false, Aa1, false, Ba1, (short)0, ca, false, false);
        v8f cb = {};
        cb = __builtin_amdgcn_wmma_f32_16x16x4_f32(false, Ab0, false, Bb0, (short)0, cb, false, false);
        cb = __builtin_amdgcn_wmma_f32_16x16x4_f32(false, Ab1, false, Bb1, (short)0, cb, false, false);

        // all rows of A identical -> every element of D equals the dot product
        const float dot  = (lane < 16) ? ca[0] : cb[0];
        const int   my_t = ta + (lane >> 4);
        const int   gt   = t0 + my_t;
        const int   gn   = n0 + nl;

        const float th0 = row[my_t * Kk + 0];
        const float eps = y[gn * Tt + gt] - th0 - dot;

        if (gt > 0) eps_out[gn * T1 + (gt - 1)] = eps;
        else        eps0[gn] = eps;

        // alpha=theta[..,9], beta=theta[..,10], omega=theta[..,11] -> [t][n] planes
        aT[gt * Nn + gn] = row[my_t * Kk + 9];
        bT[gt * Nn + gn] = row[my_t * Kk + 10];
        oT[gt * Nn + gn] = row[my_t * Kk + 11];
    }
}

// ---------------------------------------------------------------------------
// Phase 2a: per-(n, chunk) composition of the linear recurrence s = a*s + c
// ---------------------------------------------------------------------------
__global__ __launch_bounds__(256) void garch_phase2a(
    const float* __restrict__ aT, const float* __restrict__ bT,
    const float* __restrict__ oT, const float* __restrict__ eps_out,
    const float* __restrict__ eps0,
    float* __restrict__ chunkA, float* __restrict__ chunkC)
{
    const int gtid = blockIdx.x * 256 + threadIdx.x;
    const int wid  = gtid >> 5;
    const int lane = gtid & 31;
    const int ng   = wid >> 4;
    const int c    = wid & (NCH - 1);
    const int n    = ng * 32 + lane;
    const int ts   = 1 + c * CHL;
    const int te   = min(Tt, ts + CHL);

    float A = 1.0f, C = 0.0f;
    for (int t = ts; t < te; ++t) {
        const float a  = aT[t * Nn + n];
        const float b  = bT[t * Nn + n];
        const float o  = oT[t * Nn + n];
        const float ep = (t == 1) ? eps0[n] : eps_out[n * T1 + (t - 2)];
        const float ct = o + b * ep * ep;
        C = a * C + ct;
        A = a * A;
    }
    chunkA[n * NCH + c] = A;
    chunkC[n * NCH + c] = C;
}

// ---------------------------------------------------------------------------
// Phase 2b: tiny cross-chunk scan (16 steps per series)
// ---------------------------------------------------------------------------
__global__ void garch_phase2b(const float* __restrict__ oT,
                              const float* __restrict__ chunkA,
                              const float* __restrict__ chunkC,
                              float* __restrict__ chunkS)
{
    const int n = blockIdx.x * 256 + threadIdx.x;
    float s = oT[n];                       // oT[0][n] == theta[n,0,11] == s0
    for (int c = 0; c < NCH; ++c) {
        chunkS[n * NCH + c] = s;
        s = chunkA[n * NCH + c] * s + chunkC[n * NCH + c];
    }
}

// ---------------------------------------------------------------------------
// Phase 2c: expand chunks, write sigma2, deterministic loss tree-reduction
// ---------------------------------------------------------------------------
__global__ __launch_bounds__(256) void garch_phase2c(
    const float* __restrict__ aT, const float* __restrict__ bT,
    const float* __restrict__ oT, const float* __restrict__ eps_out,
    const float* __restrict__ eps0, const float* __restrict__ chunkS,
    float* __restrict__ sigma2, float* __restrict__ partials)
{
    __shared__ float red[256];
    const int tid  = threadIdx.x;
    const int gtid = blockIdx.x * 256 + tid;
    const int wid  = gtid >> 5;
    const int lane = gtid & 31;
    const int ng   = wid >> 4;
    const int c    = wid & (NCH - 1);
    const int n    = ng * 32 + lane;
    const int ts   = 1 + c * CHL;
    const int te   = min(Tt, ts + CHL);

    float s   = chunkS[n * NCH + c];
    float acc = 0.0f;
    for (int t = ts; t < te; ++t) {
        const float a  = aT[t * Nn + n];
        const float b  = bT[t * Nn + n];
        const float o  = oT[t * Nn + n];
        const float ep = (t == 1) ? eps0[n] : eps_out[n * T1 + (t - 2)];
        s = o + a * s + b * ep * ep;
        sigma2[n * T1 + (t - 1)] = s;
        const float ec = eps_out[n * T1 + (t - 1)];
        acc += ec * ec / s + logf(s);
    }
    red[tid] = acc;
    __syncthreads();
    for (int st = 128; st > 0; st >>= 1) {
        if (tid < st) red[tid] += red[tid + st];
        __syncthreads();
    }
    if (tid == 0) partials[blockIdx.x] = red[0];
}

// ---------------------------------------------------------------------------
// Phase 3: final deterministic reduction -> loss
// ---------------------------------------------------------------------------
__global__ void garch_phase3(const float* __restrict__ partials,
                             float* __restrict__ loss)
{
    __shared__ float red[128];
    const int tid = threadIdx.x;      // 128 threads
    red[tid] = partials[tid];
    __syncthreads();
    for (int st = 64; st > 0; st >>= 1) {
        if (tid < st) red[tid] += red[tid + st];
        __syncthreads();
    }
    if (tid == 0) *loss = red[0] * (1.0f / (float)((uint64_t)Nn * (uint64_t)T1));
}

// ---------------------------------------------------------------------------
extern "C" void kernel_launch(void* const* d_in, const int* in_sizes, int n_in,
                              void* d_out, int out_size, void* d_ws, size_t ws_size,
                              hipStream_t stream)
{
    (void)in_sizes; (void)n_in; (void)out_size; (void)ws_size;

    const float* y     = (const float*)d_in[0];
    const float* f     = (const float*)d_in[1];
    const float* theta = (const float*)d_in[2];

    float* out     = (float*)d_out;
    float* loss    = out;
    float* eps_out = out + 1;
    float* sigma2  = out + 1 + (size_t)Nn * T1;

    float* ws = (float*)d_ws;
    size_t off = 0;
    float* aT     = ws + off; off += (size_t)Tt * Nn;
    float* bT     = ws + off; off += (size_t)Tt * Nn;
    float* oT     = ws + off; off += (size_t)Tt * Nn;
    float* eps0   = ws + off; off += Nn;
    float* chunkA = ws + off; off += (size_t)Nn * NCH;
    float* chunkC = ws + off; off += (size_t)Nn * NCH;
    float* chunkS = ws + off; off += (size_t)Nn * NCH;
    float* partials = ws + off; off += 128;

    dim3 g1(Tt / TT_TILE, Nn / NT_TILE);   // 50 x 64
    garch_phase1<<<g1, 256, 0, stream>>>(y, f, theta, eps_out, aT, bT, oT, eps0);
    garch_phase2a<<<(Nn / 32) * NCH * 32 / 256, 256, 0, stream>>>(aT, bT, oT, eps_out, eps0, chunkA, chunkC);
    garch_phase2b<<<Nn / 256, 256, 0, stream>>>(oT, chunkA, chunkC, chunkS);
    garch_phase2c<<<(Nn / 32) * NCH * 32 / 256, 256, 0, stream>>>(aT, bT, oT, eps_out, eps0, chunkS, sigma2, partials);
    garch_phase3<<<1, 128, 0, stream>>>(partials, loss);
}